// MyLoss_1614907703898
// MI455X (gfx1250) — compile-verified
//
#include <hip/hip_runtime.h>

typedef __attribute__((ext_vector_type(2))) float v2f;
typedef __attribute__((ext_vector_type(4))) float v4f;
typedef __attribute__((ext_vector_type(8))) float v8f;
typedef __attribute__((ext_vector_type(4))) int   v4i;

#define NDIM 16384
#define KDIM 16
#define JCHUNK 128
#define PSTRIDE 132                 // LDS dwords per P-tile row (128 + 4 pad -> conflict-free)
#define WAVES_PER_BLOCK 4
#define THREADS (WAVES_PER_BLOCK * 32)
#define NSEG 4
#define JSEG (NDIM / NSEG)          // 4096 columns per block
#define NBLOCKS ((NDIM / 16 / WAVES_PER_BLOCK) * NSEG)   // 1024
#define NCHUNK (JSEG / JCHUNK)      // 32 chunks per block
#define PTILE (16 * PSTRIDE)        // dwords per wave P tile
#define INTILE (JCHUNK * KDIM)      // dwords per input chunk

#if defined(__AMDGCN__) && __has_builtin(__builtin_amdgcn_global_load_async_to_lds_b128)
#define USE_ASYNC 1
#else
#define USE_ASYNC 0
#endif

#define CPOL_RT 0   // regular temporal (input: keep resident in L2)
#define CPOL_NT 1   // th:TH_LOAD_NT (P: streamed once, don't thrash L2)

#if USE_ASYNC
template <int CPOL>
__device__ __forceinline__ void async_b128(const float* g, float* l) {
    typedef __attribute__((address_space(1))) v4i* gp_t;   // global int4*
    typedef __attribute__((address_space(3))) v4i* lp_t;   // LDS int4*
    __builtin_amdgcn_global_load_async_to_lds_b128(
        (gp_t)(float*)g, (lp_t)l, /*offset=*/0, /*cpol=*/CPOL);
}
#endif

__device__ __forceinline__ void wait_async0() {
#if defined(__AMDGCN__)
#if __has_builtin(__builtin_amdgcn_s_wait_asynccnt)
    __builtin_amdgcn_s_wait_asynccnt(0);
#else
    asm volatile("s_wait_asynccnt 0" ::: "memory");
#endif
#endif
}

__global__ __launch_bounds__(THREADS)
void trace_pi_kernel(const float* __restrict__ inp, const float* __restrict__ P,
                     float* __restrict__ partial)
{
#if USE_ASYNC
    __shared__ float sP[2 * WAVES_PER_BLOCK * PTILE];   // double-buffered P tiles
    __shared__ float sIn[2 * INTILE];                   // double-buffered input chunk
#else
    __shared__ float sP[WAVES_PER_BLOCK * PTILE];
    __shared__ float sIn[INTILE];
#endif
    __shared__ float sRed[THREADS];

    const int tid  = threadIdx.x;
    const int wave = tid >> 5;
    const int lane = tid & 31;
    const int hi   = lane >> 4;       // lane half: selects K-pair
    const int col  = lane & 15;

    const int seg   = blockIdx.x & (NSEG - 1);
    const int grp   = blockIdx.x >> 2;
    const int strip = grp * WAVES_PER_BLOCK + wave;     // 16-row strip of P
    const long i0   = (long)strip * 16;
    const int jbase = seg * JSEG;

    v8f acc = {0.f, 0.f, 0.f, 0.f, 0.f, 0.f, 0.f, 0.f};

#if USE_ASYNC
    // ---- async double-buffered pipeline: prefetch(c+1) overlaps compute(c) ----
    auto prefetch = [&](int cidx, int buf) {
        const int jg = jbase + cidx * JCHUNK;
        float* dIn = sIn + buf * INTILE;
        float* dP  = sP + buf * (WAVES_PER_BLOCK * PTILE) + wave * PTILE;
        #pragma unroll
        for (int p = 0; p < 4; ++p) {
            const int off = p * THREADS * 4 + tid * 4;
            async_b128<CPOL_RT>(inp + (long)jg * KDIM + off, dIn + off);
        }
        #pragma unroll 4
        for (int r = 0; r < 16; ++r) {
            async_b128<CPOL_NT>(P + (i0 + r) * NDIM + jg + lane * 4,
                                dP + r * PSTRIDE + lane * 4);
        }
    };

    prefetch(0, 0);
    for (int c = 0; c < NCHUNK; ++c) {
        const int buf = c & 1;
        wait_async0();          // chunk c's LDS writes complete (only 1 prefetch in flight)
        __syncthreads();        // visible to all waves; all waves done computing chunk c-1
        if (c + 1 < NCHUNK) prefetch(c + 1, buf ^ 1);

        const float* bIn = sIn + buf * INTILE;
        const float* bP  = sP + buf * (WAVES_PER_BLOCK * PTILE) + wave * PTILE;
        #pragma unroll 8
        for (int jj = 0; jj < JCHUNK; jj += 4) {
            v2f a = *(const v2f*)(bP + col * PSTRIDE + jj + 2 * hi);
            v2f b;
            b.x = bIn[(jj + 2 * hi) * KDIM + col];
            b.y = bIn[(jj + 2 * hi + 1) * KDIM + col];
            acc = __builtin_amdgcn_wmma_f32_16x16x4_f32(
                      false, a, false, b, (short)0, acc, false, false);
        }
    }
#else
    // ---- fallback: synchronous staging (round-1 proven path), P marked non-temporal ----
    float* pW = &sP[wave * PTILE];
    for (int j0 = 0; j0 < JSEG; j0 += JCHUNK) {
        const int jg = jbase + j0;
        __syncthreads();
        #pragma unroll
        for (int p = 0; p < 4; ++p) {
            const int off = p * THREADS * 4 + tid * 4;
            v4f v = *(const v4f*)(inp + (long)jg * KDIM + off);
            *(v4f*)(sIn + off) = v;
        }
        #pragma unroll 4
        for (int r = 0; r < 16; ++r) {
            v4f v = __builtin_nontemporal_load(
                        (const v4f*)(P + (i0 + r) * NDIM + jg + lane * 4));
            *(v4f*)(pW + r * PSTRIDE + lane * 4) = v;
        }
        __syncthreads();
        #pragma unroll 8
        for (int jj = 0; jj < JCHUNK; jj += 4) {
            v2f a = *(const v2f*)(pW + col * PSTRIDE + jj + 2 * hi);
            v2f b;
            b.x = sIn[(jj + 2 * hi) * KDIM + col];
            b.y = sIn[(jj + 2 * hi + 1) * KDIM + col];
            acc = __builtin_amdgcn_wmma_f32_16x16x4_f32(
                      false, a, false, b, (short)0, acc, false, false);
        }
    }
#endif

    // Trace contribution: acc VGPR v holds Pi[i0 + v + 8*hi, col]
    float t = 0.f;
    #pragma unroll
    for (int v = 0; v < 8; ++v) {
        const int m = v + 8 * hi;
        t += acc[v] * inp[(i0 + m) * KDIM + col];
    }
    sRed[tid] = t;
    __syncthreads();
    if (tid == 0) {
        float s = 0.f;
        for (int i = 0; i < THREADS; ++i) s += sRed[i];
        partial[blockIdx.x] = s;
    }
}

__global__ __launch_bounds__(1024)
void reduce_kernel(const float* __restrict__ partial, float* __restrict__ out, int n)
{
    __shared__ float s[1024];
    const int tid = threadIdx.x;
    s[tid] = (tid < n) ? partial[tid] : 0.f;
    __syncthreads();
    #pragma unroll
    for (int w = 512; w > 0; w >>= 1) {
        if (tid < w) s[tid] += s[tid + w];
        __syncthreads();
    }
    if (tid == 0) out[0] = fabsf(s[0]) / (float)KDIM;
}

extern "C" void kernel_launch(void* const* d_in, const int* in_sizes, int n_in,
                              void* d_out, int out_size, void* d_ws, size_t ws_size,
                              hipStream_t stream)
{
    const float* inp = (const float*)d_in[0];   // [16384, 16] fp32
    const float* P   = (const float*)d_in[1];   // [16384, 16384] fp32
    float* out = (float*)d_out;                 // scalar fp32
    float* ws  = (float*)d_ws;                  // >= NBLOCKS floats

    trace_pi_kernel<<<dim3(NBLOCKS), dim3(THREADS), 0, stream>>>(inp, P, ws);
    reduce_kernel<<<dim3(1), dim3(1024), 0, stream>>>(ws, out, NBLOCKS);
}